// WH_ACE_15333033246838
// MI455X (gfx1250) — compile-verified
//
#include <hip/hip_runtime.h>
#include <hip/hip_bf16.h>

#define TT      256
#define NBATCH  64
#define NCLASS  7357
#define LLEN    50
#define BLK     256
#define PERTH   29              // ceil(7357/256)
#define NROWS   (TT * NBATCH)   // 16384

typedef float v2f __attribute__((ext_vector_type(2)));
typedef float v8f __attribute__((ext_vector_type(8)));

// ---------- block reductions (wave32-aware: 8 waves per 256-thread block) ----
__device__ __forceinline__ float block_reduce_sum(float v, float* lds) {
    #pragma unroll
    for (int o = 16; o > 0; o >>= 1) v += __shfl_xor(v, o, 32);
    const int wave = threadIdx.x >> 5, lane = threadIdx.x & 31;
    if (lane == 0) lds[wave] = v;
    __syncthreads();
    if (wave == 0) {
        v = (lane < 8) ? lds[lane] : 0.0f;
        #pragma unroll
        for (int o = 4; o > 0; o >>= 1) v += __shfl_xor(v, o, 32);
        if (lane == 0) lds[0] = v;
    }
    __syncthreads();
    float r = lds[0];
    __syncthreads();            // safe reuse of lds by the next reduction
    return r;
}

__device__ __forceinline__ float block_reduce_max(float v, float* lds) {
    #pragma unroll
    for (int o = 16; o > 0; o >>= 1) v = fmaxf(v, __shfl_xor(v, o, 32));
    const int wave = threadIdx.x >> 5, lane = threadIdx.x & 31;
    if (lane == 0) lds[wave] = v;
    __syncthreads();
    if (wave == 0) {
        v = (lane < 8) ? lds[lane] : -__builtin_inff();
        #pragma unroll
        for (int o = 4; o > 0; o >>= 1) v = fmaxf(v, __shfl_xor(v, o, 32));
        if (lane == 0) lds[0] = v;
    }
    __syncthreads();
    float r = lds[0];
    __syncthreads();
    return r;
}

// ---------- kernel 1: per-batch deduped zero-class list (class 0 + target prefix)
__global__ void whace_build_lists(const int* __restrict__ target,
                                  int* __restrict__ kArr,
                                  int* __restrict__ lists) {
    const int b = threadIdx.x;
    if (b >= NBATCH) return;
    int* lst = lists + b * 64;
    int k = 0;
    lst[k++] = 0;                       // class 0 always hollow
    for (int j = 0; j < LLEN; ++j) {
        int v = target[b * LLEN + j];
        if (v == 0) break;              // stop at first blank (cumprod validity)
        bool dup = false;
        for (int i = 0; i < k; ++i) dup |= (lst[i] == v);
        if (!dup) lst[k++] = v;         // k <= 51
    }
    kArr[b] = k;
}

// ---------- kernel 2: one block per (t,b) row; single HBM pass, NT loads -----
__global__ void __launch_bounds__(BLK)
whace_row_kernel(const float* __restrict__ pred,
                 const int* __restrict__ kArr,
                 const int* __restrict__ lists,
                 float* __restrict__ partial) {
    __shared__ float lds[16];
    const int r   = blockIdx.x;         // r = t*NBATCH + b
    const int b   = r & (NBATCH - 1);
    const int tid = threadIdx.x;
    const float* __restrict__ row = pred + (size_t)r * NCLASS;

    // cache the whole row in registers (29 per thread), non-temporal stream
    float v[PERTH];
    #pragma unroll
    for (int i = 0; i < PERTH; ++i) {
        const int c = tid + i * BLK;
        v[i] = (c < NCLASS) ? __builtin_nontemporal_load(row + c)
                            : -__builtin_inff();
    }

    float m = -__builtin_inff();
    #pragma unroll
    for (int i = 0; i < PERTH; ++i) m = fmaxf(m, v[i]);
    const float M = block_reduce_max(m, lds);

    float s = 0.0f;
    #pragma unroll
    for (int i = 0; i < PERTH; ++i) s += __expf(v[i] - M);   // exp(-inf)=0 tail
    const float S = block_reduce_sum(s, lds);

    // Z: exp-sum over the <=51 hollowed classes (gather hits L2)
    float z = 0.0f;
    const int K = kArr[b];
    if (tid < K) {
        const int cls = lists[b * 64 + tid];
        z = __expf(row[cls] - M);
    }
    const float Z = block_reduce_sum(z, lds);

    if (tid == 0) partial[r] = 1.0f - Z / S;
}

// ---------- kernel 3: 16384 -> 1 via V_WMMA_F32_16X16X4_F32 (A = ones) -------
// D[i][j] = sum_k B[k][j]; every B element contributes to exactly one column
// sum, so the grand total (sum of D row 0) is layout-independent and exact f32.
__global__ void whace_final_wmma(const float* __restrict__ partial,
                                 float* __restrict__ out) {
    const int lane = threadIdx.x;       // one wave32, all lanes active (EXEC=~0)
    v8f acc = {};
    v2f ones; ones.x = 1.0f; ones.y = 1.0f;
    #pragma unroll 4
    for (int it = 0; it < NROWS / 64; ++it) {   // 256 iterations x 64 values
        const float* p = partial + it * 64 + lane * 2;
        v2f bmat; bmat.x = p[0]; bmat.y = p[1];
        acc = __builtin_amdgcn_wmma_f32_16x16x4_f32(
                  false, ones, false, bmat, (short)0, acc, false, false);
    }
    // D row 0 lives in acc[0] of lanes 0..15 (lanes 16..31 hold row 8 = same
    // column sums, so exclude them). Reduce lanes 0..15 only.
    float t = (lane < 16) ? acc[0] : 0.0f;
    #pragma unroll
    for (int o = 8; o > 0; o >>= 1) t += __shfl_xor(t, o, 32);
    if (lane == 0) out[0] = t / (float)(NBATCH * NCLASS);
}

extern "C" void kernel_launch(void* const* d_in, const int* in_sizes, int n_in,
                              void* d_out, int out_size, void* d_ws, size_t ws_size,
                              hipStream_t stream) {
    const float* pred   = (const float*)d_in[0];   // [T, NB, NCLASS] f32
    const int*   target = (const int*)d_in[1];     // [NB, L] int
    float*       out    = (float*)d_out;           // scalar

    int*   kArr    = (int*)d_ws;                   // 64 ints
    int*   lists   = kArr + 64;                    // 64 * 64 ints
    float* partial = (float*)(lists + 64 * 64);    // 16384 floats

    hipLaunchKernelGGL(whace_build_lists, dim3(1), dim3(64), 0, stream,
                       target, kArr, lists);
    hipLaunchKernelGGL(whace_row_kernel, dim3(NROWS), dim3(BLK), 0, stream,
                       pred, kArr, lists, partial);
    hipLaunchKernelGGL(whace_final_wmma, dim3(1), dim3(32), 0, stream,
                       partial, out);
}